// OrderSimilarity_7602092114141
// MI455X (gfx1250) — compile-verified
//
#include <hip/hip_runtime.h>
#include <hip/hip_bf16.h>
#include <stdint.h>

// Problem constants from the reference: lab [512,300], vis [1024,300], out [1024,512].
#define D_DIM   300
#define TILE_V  32     // vis rows per workgroup
#define TILE_L  64     // lab rows per workgroup
#define THREADS 256    // 16 (lab) x 16 (vis); each thread: 2 vis x 4 lab outputs

// gfx1250 async global->LDS copy (tracked by ASYNCcnt). VDST operand is the
// LDS byte address (wave-relative), VADDR is a 64-bit global address.
__device__ __forceinline__ void async_copy_b128(uint32_t lds_byte_addr,
                                                const float* gaddr) {
    asm volatile("global_load_async_to_lds_b128 %0, %1, off"
                 :
                 : "v"(lds_byte_addr), "v"(gaddr)
                 : "memory");
}

__device__ __forceinline__ void wait_asynccnt0() {
    asm volatile("s_wait_asynccnt 0x0" ::: "memory");
}

__global__ __launch_bounds__(THREADS)
void OrderSimilarity_kernel(const float* __restrict__ lab,
                            const float* __restrict__ vis,
                            float* __restrict__ out,
                            int n_lab, int n_vis) {
    __shared__ __align__(16) float s_lab[TILE_L * D_DIM];   // 75 KB
    __shared__ __align__(16) float s_vis[TILE_V * D_DIM];   // 37.5 KB

    const int tid = threadIdx.x;
    const int i0  = blockIdx.x * TILE_V;   // vis tile origin
    const int j0  = blockIdx.y * TILE_L;   // lab tile origin

    // ---- Stage both tiles into LDS with async b128 copies ----------------
    // Tiles are contiguous row blocks, so this is a flat copy split across
    // lanes. 300 floats/row -> every row is 16B aligned (1200 % 16 == 0).
    constexpr int LAB_VEC = TILE_L * D_DIM / 4;   // 4800 float4
    constexpr int VIS_VEC = TILE_V * D_DIM / 4;   // 2400 float4
    const float* glab = lab + (size_t)j0 * D_DIM;
    const float* gvis = vis + (size_t)i0 * D_DIM;
    for (int k = tid; k < LAB_VEC; k += THREADS)
        async_copy_b128((uint32_t)(uintptr_t)&s_lab[k * 4], glab + k * 4);
    for (int k = tid; k < VIS_VEC; k += THREADS)
        async_copy_b128((uint32_t)(uintptr_t)&s_vis[k * 4], gvis + k * 4);
    wait_asynccnt0();      // my wave's LDS writes have landed
    __syncthreads();       // everyone's writes visible

    // ---- Register-blocked relu-distance accumulation ---------------------
    const int tx = tid & 15;   // lab sub-index: cols tx + 16a, a = 0..3
    const int ty = tid >> 4;   // vis sub-index: rows ty + 16b, b = 0..1

    float acc[2][4];
#pragma unroll
    for (int b = 0; b < 2; ++b)
#pragma unroll
        for (int a = 0; a < 4; ++a) acc[b][a] = 0.0f;

    const float* plab0 = &s_lab[tx * D_DIM];
    const float* pvis0 = &s_vis[ty * D_DIM];

#pragma unroll 4
    for (int d = 0; d < D_DIM; ++d) {
        float lv[4], vv[2];
#pragma unroll
        for (int a = 0; a < 4; ++a) lv[a] = plab0[a * 16 * D_DIM + d];
#pragma unroll
        for (int b = 0; b < 2; ++b) vv[b] = pvis0[b * 16 * D_DIM + d];
#pragma unroll
        for (int b = 0; b < 2; ++b)
#pragma unroll
            for (int a = 0; a < 4; ++a) {
                float x = lv[a] - vv[b];
                x = fmaxf(x, 0.0f);
                acc[b][a] = fmaf(x, x, acc[b][a]);
            }
    }

    // ---- Epilogue: out[i, j] = -sqrt(acc) --------------------------------
#pragma unroll
    for (int b = 0; b < 2; ++b) {
        const int i = i0 + ty + 16 * b;
#pragma unroll
        for (int a = 0; a < 4; ++a) {
            const int j = j0 + tx + 16 * a;
            out[(size_t)i * n_lab + j] = -__builtin_sqrtf(acc[b][a]);
        }
    }
}

extern "C" void kernel_launch(void* const* d_in, const int* in_sizes, int n_in,
                              void* d_out, int out_size, void* d_ws, size_t ws_size,
                              hipStream_t stream) {
    const float* lab = (const float*)d_in[0];   // [n_lab, 300]
    const float* vis = (const float*)d_in[1];   // [n_vis, 300]
    float* out = (float*)d_out;                 // [n_vis, n_lab]

    const int n_lab = in_sizes[0] / D_DIM;      // 512
    const int n_vis = in_sizes[1] / D_DIM;      // 1024

    dim3 grid(n_vis / TILE_V, n_lab / TILE_L);  // (32, 8) = 256 workgroups
    OrderSimilarity_kernel<<<grid, THREADS, 0, stream>>>(lab, vis, out,
                                                         n_lab, n_vis);
}